// FraudGNN_69458211111381
// MI455X (gfx1250) — compile-verified
//
#include <hip/hip_runtime.h>
#include <hip/hip_bf16.h>

#define HID 64
#define HC 256     // HEADS*C
#define HEADS 4
#define C 64
#define G 64
#define NEG_SLOPE 0.2f

typedef float v2f __attribute__((ext_vector_type(2)));
typedef float v8f __attribute__((ext_vector_type(8)));

// -------- ordered-int float atomic max (works for any sign, init -1e30) -----
__device__ __forceinline__ void atomicMaxF(float* addr, float val) {
    if (val >= 0.0f) {
        atomicMax((int*)addr, __float_as_int(val));
    } else {
        atomicMin((unsigned int*)addr, __float_as_uint(val));
    }
}

// -------- input linear: h[n,64] = x[n,16] @ W_in^T + b_in -------------------
__global__ void k_input_linear(const float* __restrict__ x,
                               const float* __restrict__ Wi,
                               const float* __restrict__ bi,
                               float* __restrict__ h, int N) {
    int i = blockIdx.x * blockDim.x + threadIdx.x;
    if (i >= N * HID) return;
    int n = i >> 6, k = i & 63;
    const float* xr = x + (size_t)n * 16;
    const float* wr = Wi + (size_t)k * 16;
    float s = bi[k];
#pragma unroll
    for (int j = 0; j < 16; ++j) s += xr[j] * wr[j];
    h[i] = s;
}

// -------- node projection GEMM via f32 WMMA: hp[N,256] = h[N,64] @ W^T ------
// W is [256,64] row-major. One wave computes a 16x16 tile of hp, K=64 in 16
// steps of V_WMMA_F32_16X16X4_F32. N must be a multiple of 16 (50000 = 3125*16).
__global__ void k_proj_wmma(const float* __restrict__ h,
                            const float* __restrict__ W,
                            float* __restrict__ hp) {
    int lane = threadIdx.x & 31;
    int wave = (blockIdx.x * blockDim.x + threadIdx.x) >> 5;
    int rowTile = wave >> 4;          // 16 col-tiles cover 256 outputs
    int colTile = wave & 15;
    int row0 = rowTile * 16;
    int col0 = colTile * 16;
    int lrow = lane & 15;
    int hi   = lane >> 4;             // 0: K pair {0,1}; 1: K pair {2,3}

    const float* ha = h + (size_t)(row0 + lrow) * HID + 2 * hi;   // A[M=lrow, K..]
    const float* wb = W + (size_t)(col0 + lrow) * HID + 2 * hi;   // B[K.., N=lrow] = W^T

    v8f acc = {0.f, 0.f, 0.f, 0.f, 0.f, 0.f, 0.f, 0.f};
#pragma unroll
    for (int kk = 0; kk < HID; kk += 4) {
        v2f a = *(const v2f*)(ha + kk);
        v2f b = *(const v2f*)(wb + kk);
        acc = __builtin_amdgcn_wmma_f32_16x16x4_f32(
            /*neg_a=*/false, a, /*neg_b=*/false, b,
            /*c_mod=*/(short)0, acc, /*reuse_a=*/false, /*reuse_b=*/false);
    }
    // C/D layout: VGPR r -> M=r (lanes 0-15), M=r+8 (lanes 16-31); N = lane%16
    float* o = hp + (size_t)row0 * HC + col0 + lrow;
#pragma unroll
    for (int r = 0; r < 8; ++r) {
        o[(size_t)(r + 8 * hi) * HC] = acc[r];
    }
}

// -------- per (node, head) attention logits ---------------------------------
__global__ void k_alpha(const float* __restrict__ hp,
                        const float* __restrict__ asrc,
                        const float* __restrict__ adst,
                        float* __restrict__ as_, float* __restrict__ ad_, int N) {
    int i = blockIdx.x * blockDim.x + threadIdx.x;
    if (i >= N * HEADS) return;
    int n = i >> 2, hd = i & 3;
    const float* hr = hp + (size_t)n * HC + hd * C;
    const float* sa = asrc + hd * C;
    const float* da = adst + hd * C;
    float s = 0.f, d = 0.f;
#pragma unroll 8
    for (int c = 0; c < C; ++c) {
        float v = hr[c];
        s += v * sa[c];
        d += v * da[c];
    }
    as_[i] = s;
    ad_[i] = d;
}

// -------- init per-layer accumulators ---------------------------------------
__global__ void k_init_layer(float* __restrict__ agg, float* __restrict__ smax,
                             float* __restrict__ ssum, int N) {
    int i = blockIdx.x * blockDim.x + threadIdx.x;
    if (i < N * HC) agg[i] = 0.f;
    if (i < N * HEADS) { smax[i] = -1e30f; ssum[i] = 0.f; }
}

// -------- edge pass 1: segment max of leaky_relu(as[src]+ad[dst]) -----------
__global__ void k_edge_max(const int* __restrict__ ei,
                           const float* __restrict__ as_,
                           const float* __restrict__ ad_,
                           float* __restrict__ smax, int E, int Etot) {
    int e = blockIdx.x * blockDim.x + threadIdx.x;
    if (e >= Etot) return;
    int s, d;
    if (e < E) { s = ei[e]; d = ei[E + e]; } else { s = d = e - E; }
#pragma unroll
    for (int h = 0; h < HEADS; ++h) {
        float v = as_[s * HEADS + h] + ad_[d * HEADS + h];
        v = (v > 0.f) ? v : NEG_SLOPE * v;
        atomicMaxF(&smax[d * HEADS + h], v);
    }
}

// -------- edge pass 2: w = exp(e - m[dst]); segsum += w; agg[dst] += w*hp[src]
// one wave per edge; 32 lanes x 8 iters cover the 256 message channels.
__global__ void k_edge_agg(const int* __restrict__ ei,
                           const float* __restrict__ as_,
                           const float* __restrict__ ad_,
                           const float* __restrict__ smax,
                           const float* __restrict__ hp,
                           float* __restrict__ ssum,
                           float* __restrict__ agg, int E, int Etot) {
    int lane = threadIdx.x & 31;
    int eid = (blockIdx.x * blockDim.x + threadIdx.x) >> 5;
    if (eid >= Etot) return;
    int s, d;
    if (eid < E) { s = ei[eid]; d = ei[E + eid]; } else { s = d = eid - E; }

    float wl = 0.f;
    if (lane < HEADS) {
        float v = as_[s * HEADS + lane] + ad_[d * HEADS + lane];
        v = (v > 0.f) ? v : NEG_SLOPE * v;
        wl = __expf(v - smax[d * HEADS + lane]);
        atomicAdd(&ssum[d * HEADS + lane], wl);
    }
    float w0 = __shfl(wl, 0), w1 = __shfl(wl, 1), w2 = __shfl(wl, 2), w3 = __shfl(wl, 3);

    const float* hs = hp + (size_t)s * HC;
    float* ag = agg + (size_t)d * HC;
#pragma unroll
    for (int it = 0; it < 8; ++it) {
        int idx = it * 32 + lane;
        float w = ((it >> 1) == 0) ? w0 : ((it >> 1) == 1) ? w1 : ((it >> 1) == 2) ? w2 : w3;
        atomicAdd(&ag[idx], w * hs[idx]);
    }
}

// -------- node epilogue: normalize, head-mean, +bias, relu ------------------
__global__ void k_node_out(const float* __restrict__ agg,
                           const float* __restrict__ ssum,
                           const float* __restrict__ bg,
                           float* __restrict__ hout, int N) {
    int i = blockIdx.x * blockDim.x + threadIdx.x;
    if (i >= N * HID) return;
    int n = i >> 6, k = i & 63;
    float acc = 0.f;
#pragma unroll
    for (int h = 0; h < HEADS; ++h)
        acc += agg[(size_t)n * HC + h * C + k] / (ssum[n * HEADS + h] + 1e-16f);
    acc = 0.25f * acc + bg[k];
    hout[i] = (acc > 0.f) ? acc : 0.f;
}

// -------- global mean pool + final linear + sigmoid -------------------------
__global__ void k_pool_zero(float* __restrict__ psum, float* __restrict__ pcnt) {
    int i = blockIdx.x * blockDim.x + threadIdx.x;
    if (i < G * HID) psum[i] = 0.f;
    if (i < G) pcnt[i] = 0.f;
}

__global__ void k_pool_acc(const float* __restrict__ h, const int* __restrict__ batch,
                           float* __restrict__ psum, float* __restrict__ pcnt, int N) {
    int i = blockIdx.x * blockDim.x + threadIdx.x;
    if (i >= N * HID) return;
    int n = i >> 6, k = i & 63;
    int g = batch[n];
    atomicAdd(&psum[g * HID + k], h[i]);
    if (k == 0) atomicAdd(&pcnt[g], 1.f);
}

__global__ void k_final(const float* __restrict__ psum, const float* __restrict__ pcnt,
                        const float* __restrict__ Wo, const float* __restrict__ bo,
                        float* __restrict__ out) {
    int g = threadIdx.x;
    if (g >= G) return;
    float c = pcnt[g];
    c = (c > 1.f) ? c : 1.f;
    float s = bo[0];
    const float* pr = psum + g * HID;
#pragma unroll 8
    for (int k = 0; k < HID; ++k) s += (pr[k] / c) * Wo[k];
    out[g] = 1.f / (1.f + __expf(-s));
}

// ---------------------------------------------------------------------------
extern "C" void kernel_launch(void* const* d_in, const int* in_sizes, int n_in,
                              void* d_out, int out_size, void* d_ws, size_t ws_size,
                              hipStream_t stream) {
    const float* x     = (const float*)d_in[0];
    const int*   ei    = (const int*)d_in[1];     // [2,E] row-major (int32)
    const int*   batch = (const int*)d_in[2];
    const float* W_in  = (const float*)d_in[3];
    const float* b_in  = (const float*)d_in[4];
    const float* W_out = (const float*)d_in[5];
    const float* b_out = (const float*)d_in[6];

    const int N = in_sizes[0] / 16;
    const int E = in_sizes[1] / 2;
    const int Etot = E + N;           // + self loops

    // workspace layout (floats)
    float* ws   = (float*)d_ws;
    float* hA   = ws;                              // N*HID
    float* hB   = hA + (size_t)N * HID;            // N*HID
    float* hp   = hB + (size_t)N * HID;            // N*HC
    float* agg  = hp + (size_t)N * HC;             // N*HC
    float* as_  = agg + (size_t)N * HC;            // N*HEADS
    float* ad_  = as_ + (size_t)N * HEADS;         // N*HEADS
    float* smax = ad_ + (size_t)N * HEADS;         // N*HEADS
    float* ssum = smax + (size_t)N * HEADS;        // N*HEADS
    float* psum = ssum + (size_t)N * HEADS;        // G*HID
    float* pcnt = psum + (size_t)G * HID;          // G

    const int TB = 256;

    // input linear
    k_input_linear<<<(N * HID + TB - 1) / TB, TB, 0, stream>>>(x, W_in, b_in, hA, N);

    float* hin  = hA;
    float* hout = hB;
    for (int l = 0; l < 3; ++l) {
        const float* W    = (const float*)d_in[7 + 4 * l];
        const float* asrc = (const float*)d_in[8 + 4 * l];
        const float* adst = (const float*)d_in[9 + 4 * l];
        const float* bg   = (const float*)d_in[10 + 4 * l];

        // projection GEMM (WMMA): (N/16)*16 waves, 8 waves / 256-thread block
        int waves = (N / 16) * 16;
        k_proj_wmma<<<waves / 8, TB, 0, stream>>>(hin, W, hp);

        k_alpha<<<(N * HEADS + TB - 1) / TB, TB, 0, stream>>>(hp, asrc, adst, as_, ad_, N);

        k_init_layer<<<((size_t)N * HC + TB - 1) / TB, TB, 0, stream>>>(agg, smax, ssum, N);

        k_edge_max<<<(Etot + TB - 1) / TB, TB, 0, stream>>>(ei, as_, ad_, smax, E, Etot);

        // one wave per edge
        long long lanes = (long long)Etot * 32;
        k_edge_agg<<<(int)((lanes + TB - 1) / TB), TB, 0, stream>>>(
            ei, as_, ad_, smax, hp, ssum, agg, E, Etot);

        k_node_out<<<(N * HID + TB - 1) / TB, TB, 0, stream>>>(agg, ssum, bg, hout, N);

        float* t = hin; hin = hout; hout = t;   // ping-pong (result now in hin)
    }

    // global mean pool + head
    k_pool_zero<<<(G * HID + TB - 1) / TB, TB, 0, stream>>>(psum, pcnt);
    k_pool_acc<<<(N * HID + TB - 1) / TB, TB, 0, stream>>>(hin, batch, psum, pcnt, N);
    k_final<<<1, 64, 0, stream>>>(psum, pcnt, W_out, b_out, (float*)d_out);
}